// FullReplicateGenerator_25503515804223
// MI455X (gfx1250) — compile-verified
//
#include <hip/hip_runtime.h>
#include <hip/hip_bf16.h>
#include <stdint.h>

// FullReplicateGenerator for MI455X (gfx1250, wave32).
// One wave owns 16 (b,t) rows (R=4 each); 50-step diffusion chain fully in
// registers. Per-step 4x4 GEMM runs on the matrix pipe via
// V_WMMA_F32_16X16X4_F32 with A' = W^T placed in rows {0,1,8,9} so that the
// fixed C/D layout (lanes 0-15: Mt=v, lanes 16-31: Mt=v+8) lands noise_pred
// for each lane's two replicate slots directly in d[0],d[1] -- no cross-lane
// traffic at all. Replicate masks are baked into A' (wave-uniform b).
// Memory floor: one 210 MB streamed output write (~9us @ 23.3 TB/s).

typedef float v2f __attribute__((ext_vector_type(2)));
typedef float v8f __attribute__((ext_vector_type(8)));

#define B_N 65536
#define T_N 200
#define R_N 4
#define NSTEPS 50
#define NTIME 1000
#define TW 13          // ceil(T_N / 16) wave-tiles of rows per batch element

// ---------------------------------------------------------------------------
// Counter-based RNG (3-round Philox-style 32x2 mix) + Box-Muller using raw
// hw transcendentals: v_log_f32 is log2 (inputs bounded away from denorms),
// v_sin/v_cos take revolutions so u2 is passed directly (angle = 2*pi*u2).
// Pure function of (k1,k2) -> deterministic per call.
// ---------------------------------------------------------------------------
__device__ __forceinline__ void gauss2(uint32_t k1, uint32_t k2,
                                       float& z0, float& z1) {
  uint32_t c0 = k1;
  uint32_t c1 = k2 ^ 0x85EBCA6Bu;
  uint32_t key = 0xA511E9B3u;
#pragma unroll
  for (int r = 0; r < 3; ++r) {
    uint32_t lo = c0 * 0xD256D193u;
    uint32_t hi = __umulhi(c0, 0xD256D193u);
    c0 = hi ^ key ^ c1;
    c1 = lo;
    key += 0x9E3779B9u;
  }
  float u1 = (float)(c0 >> 8) * 0x1p-24f + 0x1p-25f;   // [2^-25, 1]: no denorm
  float u2 = (float)(c1 >> 8) * 0x1p-24f;              // [0, 1) revolutions
  // -2*ln(u1) = log2(u1) * (-2*ln2); log2(u1) <= 0 so product >= 0.
  float r2 = __builtin_amdgcn_logf(u1) * -1.3862943611f;
  float rr = __builtin_amdgcn_sqrtf(r2);
  z0 = rr * __builtin_amdgcn_cosf(u2);
  z1 = rr * __builtin_amdgcn_sinf(u2);
}

// ---------------------------------------------------------------------------
// Step-constant table: {rsa = 1/sqrt(a), coeff = sqrt(1-a)*ct/sqrt(a),
// sb = sqrt(beta)*0.1, pad}. Single thread, on-stream, capture-safe.
// ---------------------------------------------------------------------------
__global__ void tables_kernel(float* ws) {
  if (threadIdx.x != 0 || blockIdx.x != 0) return;
  float* tab = ws;            // 50*4 floats
  float* acp = ws + 256;      // 1000 floats scratch
  float cp = 1.0f;
  for (int t = 0; t < NTIME; ++t) {
    float beta = 0.0001f + (0.02f - 0.0001f) * ((float)t / (float)(NTIME - 1));
    cp *= (1.0f - beta);
    acp[t] = cp;
  }
  for (int i = 0; i < NSTEPS; ++i) {
    float tv = 999.0f - (float)i * (999.0f / 49.0f);
    int t = (int)(tv + 1e-3f);                      // astype(int32) truncation
    float a = fmaxf(acp[t], 1e-6f);
    float beta = 0.0001f + (0.02f - 0.0001f) * ((float)t / (float)(NTIME - 1));
    float sa  = sqrtf(a);
    float s1  = sqrtf(fmaxf(1.0f - a, 1e-6f));
    float ct  = 1.0f - (float)t / (float)NTIME;
    tab[i * 4 + 0] = 1.0f / sa;                     // rsa
    tab[i * 4 + 1] = s1 * ct / sa;                  // coeff (folds ct scale)
    tab[i * 4 + 2] = sqrtf(beta) * 0.1f;            // sb (folds 0.1 noise)
    tab[i * 4 + 3] = 0.0f;
  }
}

// ---------------------------------------------------------------------------
// Main fused kernel. 256 threads = 8 waves per block; each wave: one b and a
// 16-row tile of t. Lane l holds state x[row=l%16][k0], x[row][k0+1] with
// k0 = (l<16 ? 0 : 2) == the WMMA B-operand (4x16 f32) layout.
// ---------------------------------------------------------------------------
__global__ __launch_bounds__(256) void diffuse_kernel(
    const int* __restrict__ drug_ids, const float* __restrict__ conc,
    const int* __restrict__ repc, const float* __restrict__ drug_emb,
    const float* __restrict__ unet_W, const float* __restrict__ tab,
    float* __restrict__ out) {
  const int lane = threadIdx.x & 31;
  const int wave = (int)((blockIdx.x * blockDim.x + threadIdx.x) >> 5);
  const int b = wave / TW;
  const int t0 = (wave - b * TW) * 16;
  const int m = lane & 15;            // row index within the 16-row tile
  const bool hi = lane >= 16;
  const int t = t0 + m;               // may reach 207 on last tile; store guarded
  const int k0 = hi ? 2 : 0;          // replicate slots owned by this lane

  const int rc = repc[b];
  const float mk0 = (k0 < rc) ? 1.0f : 0.0f;
  const float mk1 = (k0 + 1 < rc) ? 1.0f : 0.0f;

  // A' operand (16x4): rows {0,1,8,9} = W^T rows {0,1,2,3}, others zero.
  // With D-layout Mt = v + 8*hi, every lane then reads np for its own slots
  // from d[0], d[1] in-lane. Replicate mask (wave-uniform b, K-slot only)
  // is folded into A' columns: masked x entries contribute nothing.
  int c = (m == 0) ? 0 : (m == 1) ? 1 : (m == 8) ? 2 : (m == 9) ? 3 : -1;
  v2f a;
  a.x = (c >= 0) ? unet_W[k0 * R_N + c] * mk0 : 0.0f;        // A'[m][k0]
  a.y = (c >= 0) ? unet_W[(k0 + 1) * R_N + c] * mk1 : 0.0f;  // A'[m][k0+1]

  // Unique RNG row key: t < 256 so (b<<8)|t never collides across b.
  const uint32_t rowkey = ((uint32_t)b << 8) | (uint32_t)t;

  // x0 = N(0,1); mask handled by A' (matmul) and by the epilogue multiply.
  v2f x;
  {
    float z0, z1;
    gauss2(rowkey, 2u * 50u + (hi ? 1u : 0u), z0, z1);
    x.x = z0;
    x.y = z1;
  }

  // ---- 50-step reverse diffusion; EXEC uniform throughout ----
  for (int i = 0; i < NSTEPS; ++i) {
    const float rsa   = tab[i * 4 + 0];
    const float coeff = tab[i * 4 + 1];
    const float sb    = tab[i * 4 + 2];

    // D = A'(16x4) x B(x^T, 4x16); np for this lane's slots = d[0], d[1].
    v8f czero = {0.f, 0.f, 0.f, 0.f, 0.f, 0.f, 0.f, 0.f};
    v8f d = __builtin_amdgcn_wmma_f32_16x16x4_f32(
        false, a, false, x, (short)0, czero, false, false);

    float p0 = fminf(fmaxf(x.x * rsa - coeff * d[0], -10.0f), 10.0f);
    float p1 = fminf(fmaxf(x.y * rsa - coeff * d[1], -10.0f), 10.0f);

    if (i < NSTEPS - 1) {            // uniform branch
      float z0, z1;
      gauss2(rowkey, 2u * (uint32_t)i + (hi ? 1u : 0u), z0, z1);
      p0 += sb * z0;
      p1 += sb * z1;
    }
    x.x = p0;
    x.y = p1;
  }

  // ---- epilogue: 30 + 10*tanh(x) + (drug_response + rep_var)*mask, *mask ----
  // tanh via raw exp2 + rcp: tanh(x) = (e-1)/(e+1), e = 2^(2x*log2e).
  float e0 = __builtin_amdgcn_exp2f(x.x * 2.8853900818f);
  float e1 = __builtin_amdgcn_exp2f(x.y * 2.8853900818f);
  float xs0 = 30.0f + 10.0f * ((e0 - 1.0f) * __builtin_amdgcn_rcpf(e0 + 1.0f));
  float xs1 = 30.0f + 10.0f * ((e1 - 1.0f) * __builtin_amdgcn_rcpf(e1 + 1.0f));

  const int did = drug_ids[b];
  const int tc = (t < T_N) ? t : (T_N - 1);
  const float dresp = drug_emb[did * T_N + tc] * conc[b];

  float z0, z1;
  gauss2(rowkey, 2u * 51u + (hi ? 1u : 0u), z0, z1);
  xs0 = (xs0 + (dresp + 0.5f * z0) * mk0) * mk0;
  xs1 = (xs1 + (dresp + 0.5f * z1) * mk1) * mk1;

  if (t < T_N) {
    // (b*T + t)*R + k0 : 8B per lane; a wave covers 256 contiguous bytes.
    size_t idx = ((size_t)b * T_N + (size_t)t) * R_N + (size_t)k0;
    float2 v2 = make_float2(xs0, xs1);
    *(float2*)(out + idx) = v2;
  }
  // masks output (B,R) appended after x_scaled
  if (t0 == 0 && lane < R_N) {
    out[(size_t)B_N * T_N * R_N + (size_t)b * R_N + lane] =
        (lane < rc) ? 1.0f : 0.0f;
  }
}

extern "C" void kernel_launch(void* const* d_in, const int* in_sizes, int n_in,
                              void* d_out, int out_size, void* d_ws,
                              size_t ws_size, hipStream_t stream) {
  (void)in_sizes; (void)n_in; (void)out_size; (void)ws_size;
  const int*   drug_ids = (const int*)d_in[0];
  const float* conc     = (const float*)d_in[1];
  const int*   repc     = (const int*)d_in[2];
  // d_in[3] = media_change_times: unused by the reference math.
  const float* drug_emb = (const float*)d_in[4];
  const float* unet_W   = (const float*)d_in[5];
  float* tab = (float*)d_ws;

  tables_kernel<<<1, 32, 0, stream>>>(tab);

  const int total_waves = B_N * TW;          // 851,968
  const int blocks = total_waves / 8;        // 8 waves (256 thr) per block
  diffuse_kernel<<<blocks, 256, 0, stream>>>(drug_ids, conc, repc, drug_emb,
                                             unet_W, tab, (float*)d_out);
}